// CircleLoss_47064251630174
// MI455X (gfx1250) — compile-verified
//
#include <hip/hip_runtime.h>
#include <math.h>

typedef __attribute__((ext_vector_type(2))) float v2f;
typedef __attribute__((ext_vector_type(8))) float v8f;

#define CL_GAMMA 256.0f
#define CL_O_P   1.25f
#define CL_O_N   (-0.25f)
#define CL_D_P   0.75f
#define CL_D_N   0.25f

static constexpr int D  = 128;     // feature dim
static constexpr int KC = D / 4;   // 32 K-chunks of 4 for f32 WMMA

// online logsumexp push: state (m, s) <- value v
__device__ __forceinline__ void lse_push(float& m, float& s, float v) {
  float nm = fmaxf(m, v);
  s = s * __expf(m - nm) + __expf(v - nm);
  m = nm;
}
// merge two lse states; (-1e30, 0) is the identity
__device__ __forceinline__ void lse_merge(float& m, float& s, float om, float os) {
  float nm = fmaxf(m, om);
  s = s * __expf(m - nm) + os * __expf(om - nm);
  m = nm;
}

// ---------------------------------------------------------------------------
// Kernel 1: inverse row norms  invn[i] = 1 / max(||x_i||, 1e-12)
// one wave per row; each lane squares 4 floats, butterfly reduce
// ---------------------------------------------------------------------------
__global__ __launch_bounds__(256) void cl_invnorm(const float* __restrict__ x,
                                                  float* __restrict__ invn, int n) {
  const int lane = threadIdx.x & 31;
  const int wave = threadIdx.x >> 5;
  const int row  = blockIdx.x * 8 + wave;
  if (row >= n) return;
  const float4 v = *reinterpret_cast<const float4*>(x + (size_t)row * D + lane * 4);
  float ss = v.x * v.x + v.y * v.y + v.z * v.z + v.w * v.w;
#pragma unroll
  for (int msk = 16; msk >= 1; msk >>= 1) ss += __shfl_xor(ss, msk, 32);
  if (lane == 0) invn[row] = 1.0f / fmaxf(sqrtf(ss), 1e-12f);
}

// ---------------------------------------------------------------------------
// Kernel 2: fused  X X^T  (f32 WMMA 16x16x4)  + circle-loss + streaming LSE
// block = 8 waves, one 16-row tile per block; waves stride over column tiles
// ---------------------------------------------------------------------------
__global__ __launch_bounds__(256) void circle_main(
    const float* __restrict__ x, const int* __restrict__ tg,
    const float* __restrict__ invn, float* __restrict__ accum, int n)
{
  __shared__ float smp[8][16], ssp[8][16], smn[8][16], ssn[8][16];

  const int lane = threadIdx.x & 31;
  const int wave = threadIdx.x >> 5;
  const int hi   = lane >> 4;      // half-wave select (K pair / row group)
  const int lo   = lane & 15;      // M index for A, N index for B/D
  const int rowBase = blockIdx.x * 16;

  // ---- preload A fragments for this block's 16 rows (stay in VGPRs) ----
  // A 16x4 f32 layout: lane L -> M = L&15; VGPR0/1 = K0+2*hi, K0+2*hi+1
  const int arow = rowBase + lo;
  v2f afrag[KC];
#pragma unroll
  for (int kc = 0; kc < KC; ++kc)
    afrag[kc] = *reinterpret_cast<const v2f*>(x + (size_t)arow * D + kc * 4 + 2 * hi);

  // per-register-row constants: D-tile VGPR r holds row M = r + 8*hi
  float invr[8]; int trow[8];
#pragma unroll
  for (int r = 0; r < 8; ++r) {
    const int row = rowBase + r + 8 * hi;
    invr[r] = invn[row];
    trow[r] = tg[row];
  }

  // per-lane partial LSE state over this lane's column slice
  float mp[8], sp[8], mn[8], sn[8];
#pragma unroll
  for (int r = 0; r < 8; ++r) { mp[r] = -1e30f; sp[r] = 0.0f; mn[r] = -1e30f; sn[r] = 0.0f; }

  const int nTiles = n >> 4;
  for (int t = wave; t < nTiles; t += 8) {
    const int col = (t << 4) + lo;                 // this lane's column
    const float* __restrict__ bp = x + (size_t)col * D + 2 * hi;
    v8f acc = {};
#pragma unroll
    for (int kc = 0; kc < KC; ++kc) {
      const v2f b = *reinterpret_cast<const v2f*>(bp + kc * 4);
      // D = A(16x4) * B(4x16) + C, f32 everywhere
      acc = __builtin_amdgcn_wmma_f32_16x16x4_f32(false, afrag[kc], false, b,
                                                  (short)0, acc, false, false);
    }
    const float invc = invn[col];
    const int   tcol = tg[col];
#pragma unroll
    for (int r = 0; r < 8; ++r) {
      const float v   = acc[r] * invr[r] * invc;   // cosine similarity
      const int   row = rowBase + r + 8 * hi;
      if (trow[r] == tcol) {
        if (row != col) {                          // positive (exclude self)
          const float p = -CL_GAMMA * fmaxf(CL_O_P - v, 0.0f) * (v - CL_D_P);
          lse_push(mp[r], sp[r], p);
        }
      } else {                                     // negative
        const float q = CL_GAMMA * fmaxf(v - CL_O_N, 0.0f) * (v - CL_D_N);
        lse_push(mn[r], sn[r], q);
      }
    }
  }

  // ---- fold the 16 column-lanes of each half-wave (butterfly, width 16) ----
#pragma unroll
  for (int r = 0; r < 8; ++r) {
#pragma unroll
    for (int msk = 8; msk >= 1; msk >>= 1) {
      lse_merge(mp[r], sp[r], __shfl_xor(mp[r], msk, 16), __shfl_xor(sp[r], msk, 16));
      lse_merge(mn[r], sn[r], __shfl_xor(mn[r], msk, 16), __shfl_xor(sn[r], msk, 16));
    }
  }

  // ---- publish per-wave per-row state to LDS, merge across the 8 waves ----
  if (lo == 0) {
#pragma unroll
    for (int r = 0; r < 8; ++r) {
      const int ridx = r + 8 * hi;
      smp[wave][ridx] = mp[r]; ssp[wave][ridx] = sp[r];
      smn[wave][ridx] = mn[r]; ssn[wave][ridx] = sn[r];
    }
  }
  __syncthreads();

  if (wave == 0) {
    const int ridx = lane & 15;                    // lanes 16-31 duplicate, masked below
    float Mp = -1e30f, Sp = 0.0f, Mn = -1e30f, Sn = 0.0f;
#pragma unroll
    for (int w = 0; w < 8; ++w) {
      lse_merge(Mp, Sp, smp[w][ridx], ssp[w][ridx]);
      lse_merge(Mn, Sn, smn[w][ridx], ssn[w][ridx]);
    }
    const bool valid = (Sp > 0.0f) && (Sn > 0.0f) && (lane < 16);
    const float z = (Mp + logf(Sp)) + (Mn + logf(Sn));
    const float spl = (z > 20.0f) ? z
                     : ((z < -20.0f) ? __expf(z) : log1pf(__expf(z)));
    float lsum = valid ? spl : 0.0f;
    float lcnt = valid ? 1.0f : 0.0f;
#pragma unroll
    for (int msk = 16; msk >= 1; msk >>= 1) {
      lsum += __shfl_xor(lsum, msk, 32);
      lcnt += __shfl_xor(lcnt, msk, 32);
    }
    if (lane == 0) {
      atomicAdd(accum + 0, lsum);
      atomicAdd(accum + 1, lcnt);
    }
  }
}

// ---------------------------------------------------------------------------
__global__ void cl_init(float* __restrict__ accum) {
  accum[0] = 0.0f;
  accum[1] = 0.0f;
}

__global__ void cl_finalize(const float* __restrict__ accum, float* __restrict__ out) {
  out[0] = accum[0] / fmaxf(accum[1], 1.0f);
}

// ---------------------------------------------------------------------------
extern "C" void kernel_launch(void* const* d_in, const int* in_sizes, int n_in,
                              void* d_out, int out_size, void* d_ws, size_t ws_size,
                              hipStream_t stream) {
  (void)n_in; (void)out_size; (void)ws_size;
  const float* x  = (const float*)d_in[0];   // [n, 128] f32
  const int*   tg = (const int*)d_in[1];     // [n] labels
  float* out = (float*)d_out;                // scalar f32

  const int n = in_sizes[1];                 // 8192

  float* wsf   = (float*)d_ws;
  float* accum = wsf;                        // 2 floats: sum, valid count
  float* invn  = wsf + 16;                   // n floats of inverse norms

  cl_init<<<1, 1, 0, stream>>>(accum);
  cl_invnorm<<<(n + 7) / 8, 256, 0, stream>>>(x, invn, n);
  circle_main<<<n / 16, 256, 0, stream>>>(x, tg, invn, accum, n);
  cl_finalize<<<1, 1, 0, stream>>>(accum, out);
}